// GPR_sparse_32126355374958
// MI455X (gfx1250) — compile-verified
//
#include <hip/hip_runtime.h>

#define NN 100000
#define NE 600000
#define HID 128
#define NLAY 4

typedef float v2f __attribute__((ext_vector_type(2)));
typedef float v8f __attribute__((ext_vector_type(8)));

// ---------------------------------------------------------------------------
// Dense projection H = X @ W + b using V_WMMA_F32_16X16X4_F32 (exact f32).
// Block: 256 threads = 8 waves; covers 32 rows x 128 cols of H.
// Each wave: one 16-col slice (n0 = wave*16), two 16-row tiles (B frags reused).
// A-frag layout (ISA 7.12.2, 32-bit A 16x4): lane&15 = M row, half-wave picks
//   K pair {k0,k0+1} (lanes 0-15) vs {k0+2,k0+3} (lanes 16-31).
// B-frag mirrored: VGPR0/1 hold K rows {k0,k0+1}|{k0+2,k0+3}, N = n0 + (lane&15).
// D layout: vgpr j -> row m0 + half*8 + j, col n0 + (lane&15).
// ---------------------------------------------------------------------------
__global__ __launch_bounds__(256)
void gcn_gemm_wmma(const float* __restrict__ X, const float* __restrict__ W,
                   const float* __restrict__ bias, float* __restrict__ H) {
  __shared__ float Xs[32 * 130];           // padded stride: kill LDS bank conflicts
  const int t = threadIdx.x;
  const int row0 = blockIdx.x * 32;

  // Stage 32x128 X tile (coalesced 256-thread copy).
#pragma unroll
  for (int i = 0; i < 16; ++i) {
    const int flat = t + i * 256;
    const int r = flat >> 7, c = flat & 127;
    Xs[r * 130 + c] = X[(size_t)(row0 + r) * HID + c];
  }
  __syncthreads();

  const int lane = t & 31;
  const int wave = t >> 5;
  const int r = lane & 15;
  const int half = lane >> 4;
  const int n0 = wave * 16;

  v8f d0 = {0.f, 0.f, 0.f, 0.f, 0.f, 0.f, 0.f, 0.f};
  v8f d1 = {0.f, 0.f, 0.f, 0.f, 0.f, 0.f, 0.f, 0.f};

#pragma unroll 4
  for (int k0 = 0; k0 < HID; k0 += 4) {
    const int ka = k0 + half * 2;
    v2f b;
    b.x = W[ka * HID + n0 + r];
    b.y = W[(ka + 1) * HID + n0 + r];
    v2f a0, a1;
    a0.x = Xs[r * 130 + ka];
    a0.y = Xs[r * 130 + ka + 1];
    a1.x = Xs[(r + 16) * 130 + ka];
    a1.y = Xs[(r + 16) * 130 + ka + 1];
    d0 = __builtin_amdgcn_wmma_f32_16x16x4_f32(false, a0, false, b, (short)0, d0,
                                               false, false);
    d1 = __builtin_amdgcn_wmma_f32_16x16x4_f32(false, a1, false, b, (short)0, d1,
                                               false, false);
  }

  const float bv = bias[n0 + r];
#pragma unroll
  for (int j = 0; j < 8; ++j) {
    H[(size_t)(row0 + half * 8 + j) * HID + n0 + r] = d0[j] + bv;
    H[(size_t)(row0 + 16 + half * 8 + j) * HID + n0 + r] = d1[j] + bv;
  }
}

// ---------------------------------------------------------------------------
// Edge scatter: acc[dst[e], :] += h[src[e], :] * w[e].
// 32 lanes per edge, float4 per lane (coalesced 512B row gather),
// 4x global_atomic_add_f32 per lane. e is wave-uniform -> scalar index loads.
// ---------------------------------------------------------------------------
__global__ __launch_bounds__(256)
void edge_scatter(const float* __restrict__ h, const int* __restrict__ src,
                  const int* __restrict__ dst, const float* __restrict__ w,
                  float* __restrict__ acc) {
  const int gid = blockIdx.x * 256 + threadIdx.x;
  const int e = gid >> 5;
  if (e >= NE) return;
  const int s = src[e];
  const int d = dst[e];
  const float wv = w[e];
  const int c = (gid & 31) * 4;

  const float4 v = *reinterpret_cast<const float4*>(h + (size_t)s * HID + c);
  float* base = acc + (size_t)d * HID + c;
  atomicAdd(base + 0, v.x * wv);
  atomicAdd(base + 1, v.y * wv);
  atomicAdd(base + 2, v.z * wv);
  atomicAdd(base + 3, v.w * wv);
}

// ---------------------------------------------------------------------------
// Fused activation + energy tap. One wave per node.
//   v = do_relu ? leaky_relu(in[node,:]) : in[node,:]
//   if (xout) xout[node,:] = v
//   tap = (dot(v, We_row) + be) * temp ; energy[node] {=, +=} tap
// Wave32 reduction via __shfl_xor.
// ---------------------------------------------------------------------------
__global__ __launch_bounds__(256)
void act_energy(const float* __restrict__ in, float* __restrict__ xout,
                float* __restrict__ energy, const float* __restrict__ We_row,
                const float* __restrict__ be_p, const float* __restrict__ temp_p,
                int do_relu, int accumulate) {
  const int gid = blockIdx.x * 256 + threadIdx.x;
  const int node = gid >> 5;
  if (node >= NN) return;
  const int c = (gid & 31) * 4;

  float4 v = *reinterpret_cast<const float4*>(in + (size_t)node * HID + c);
  if (do_relu) {
    v.x = fmaxf(v.x, 0.f) + 0.01f * fminf(v.x, 0.f);
    v.y = fmaxf(v.y, 0.f) + 0.01f * fminf(v.y, 0.f);
    v.z = fmaxf(v.z, 0.f) + 0.01f * fminf(v.z, 0.f);
    v.w = fmaxf(v.w, 0.f) + 0.01f * fminf(v.w, 0.f);
  }
  if (xout) *reinterpret_cast<float4*>(xout + (size_t)node * HID + c) = v;

  const float4 we = *reinterpret_cast<const float4*>(We_row + c);
  float p = v.x * we.x + v.y * we.y + v.z * we.z + v.w * we.w;
#pragma unroll
  for (int off = 16; off > 0; off >>= 1) p += __shfl_xor(p, off, 32);

  if ((gid & 31) == 0) {
    const float val = (p + be_p[0]) * temp_p[0];
    if (accumulate)
      energy[node] += val;
    else
      energy[node] = val;
  }
}

extern "C" void kernel_launch(void* const* d_in, const int* in_sizes, int n_in,
                              void* d_out, int out_size, void* d_ws, size_t ws_size,
                              hipStream_t stream) {
  const float* x    = (const float*)d_in[0];   // [NN, HID]
  const int*   src  = (const int*)d_in[1];     // [NE]
  const int*   dst  = (const int*)d_in[2];     // [NE]
  const float* w    = (const float*)d_in[3];   // [NE]
  const float* Ws   = (const float*)d_in[4];   // [NLAY, HID, HID]
  const float* bs   = (const float*)d_in[5];   // [NLAY, HID]
  const float* We   = (const float*)d_in[6];   // [NLAY+1, HID]
  const float* be   = (const float*)d_in[7];   // [NLAY+1]
  const float* temp = (const float*)d_in[8];   // [NLAY+1]

  float* energy = (float*)d_out;               // [NN]
  float* x_out  = (float*)d_out + NN;          // [NN, HID]

  float* h    = (float*)d_ws;                  // [NN, HID] projected features
  float* accA = h + (size_t)NN * HID;          // [NN, HID] scatter accumulator A
  float* accB = accA + (size_t)NN * HID;       // [NN, HID] scatter accumulator B

  const int gemm_blocks = NN / 32;             // 3125
  const int edge_blocks = (NE * 32) / 256;     // 75000
  const int node_blocks = (NN * 32 + 255) / 256; // 12500

  // energy = (x @ We[0] + be[0]) * temp[0]
  act_energy<<<node_blocks, 256, 0, stream>>>(x, nullptr, energy, We, be, temp,
                                              /*do_relu=*/0, /*accumulate=*/0);

  const float* xcur = x;
  for (int layer = 0; layer < NLAY; ++layer) {
    gcn_gemm_wmma<<<gemm_blocks, 256, 0, stream>>>(
        xcur, Ws + (size_t)layer * HID * HID, bs + (size_t)layer * HID, h);

    float* acc = (layer & 1) ? accB : accA;
    hipMemsetAsync(acc, 0, (size_t)NN * HID * sizeof(float), stream);
    edge_scatter<<<edge_blocks, 256, 0, stream>>>(h, src, dst, w, acc);

    float* xnext = (layer == NLAY - 1) ? x_out : acc;  // in-place relu is safe
    act_energy<<<node_blocks, 256, 0, stream>>>(
        acc, xnext, energy, We + (size_t)(layer + 1) * HID, be + (layer + 1),
        temp + (layer + 1), /*do_relu=*/1, /*accumulate=*/1);
    xcur = xnext;
  }
}